// DecodeDetectionsFast_21990232556249
// MI455X (gfx1250) — compile-verified
//
#include <hip/hip_runtime.h>

// ---- problem constants (from the reference) ----
#define B_      32
#define N_      24564
#define C_      81          // n_classes = 93 - 12
#define F_      93
#define TOPK_   200
#define NEG_    (-1.0e10f)
#define IOU_T   0.45f
#define CONF_T  0.01f

#define AS1 __attribute__((address_space(1)))
#define AS3 __attribute__((address_space(3)))

typedef int v4i_ __attribute__((ext_vector_type(4)));

// ---- CDNA5 async global->LDS copy (ASYNCcnt path) ----
__device__ __forceinline__ void async_cp16(void* lds_dst, const void* gsrc) {
#if __has_builtin(__builtin_amdgcn_global_load_async_to_lds_b128)
  __builtin_amdgcn_global_load_async_to_lds_b128(
      (AS1 v4i_*)gsrc, (AS3 v4i_*)lds_dst, 0, 0);
#else
  unsigned ldsoff = (unsigned)(unsigned long long)lds_dst; // low 32b of generic LDS ptr = LDS offset
  unsigned long long ga = (unsigned long long)gsrc;
  asm volatile("global_load_async_to_lds_b128 %0, %1, off"
               :: "v"(ldsoff), "v"(ga) : "memory");
#endif
}

__device__ __forceinline__ void wait_async0() {
#if __has_builtin(__builtin_amdgcn_s_wait_asynccnt)
  __builtin_amdgcn_s_wait_asynccnt(0);
#else
  asm volatile("s_wait_asynccnt 0" ::: "memory");
#endif
}

// =====================================================================
// Kernel 1: decode. One block = 256 anchors of one batch image.
// Whole 256x93 f32 tile (95232 B, 16B-aligned since 24564*372 % 16 == 0)
// is async-copied to LDS with b128 chunks, then each thread does the
// 81-way argmax + box decode from LDS (stride-93 reads: bank-conflict-free).
// =====================================================================
__global__ __launch_bounds__(256)
void decode_kernel(const float* __restrict__ y,
                   float* __restrict__ scores,   // [B_*N_]
                   float* __restrict__ boxes,    // [B_*N_*4] as float4
                   float* __restrict__ meta) {   // [B_*N_*2] (class_id, conf)
  extern __shared__ unsigned char smem[];
  float* tile = (float*)smem;

  const int tilesPerBatch = (N_ + 255) / 256;   // 96
  const int b   = blockIdx.x / tilesPerBatch;
  const int t   = blockIdx.x % tilesPerBatch;
  const int a0  = t * 256;
  int cnt = N_ - a0; if (cnt > 256) cnt = 256;

  const unsigned char* src =
      (const unsigned char*)(y + ((size_t)b * N_ + a0) * F_);
  const int nch = (cnt * (F_ * 4)) >> 4;        // 16-byte chunks (exact: cnt%4==0)
  for (int c = threadIdx.x; c < nch; c += 256)
    async_cp16(smem + (c << 4), src + (c << 4));
  wait_async0();
  __syncthreads();

  const int tid = threadIdx.x;
  if (tid < cnt) {
    const float* row = tile + tid * F_;
    // first-max argmax over classes (matches jnp.argmax tie rule)
    float best = row[0]; int bi = 0;
    for (int c = 1; c < C_; ++c) {
      float v = row[c];
      if (v > best) { best = v; bi = c; }
    }
    // box decode: offsets -12..-1 -> 81..92
    float cx = row[81] * row[89] * row[87] + row[85];
    float cy = row[82] * row[90] * row[88] + row[86];
    float w  = expf(row[83] * row[91]) * row[87];
    float h  = expf(row[84] * row[92]) * row[88];
    float xmin = (cx - 0.5f * w) * 512.0f;
    float ymin = (cy - 0.5f * h) * 512.0f;
    float xmax = (cx + 0.5f * w) * 512.0f;
    float ymax = (cy + 0.5f * h) * 512.0f;

    size_t g = (size_t)b * N_ + a0 + tid;
    bool valid = (bi != 0) && (best > CONF_T);
    scores[g] = valid ? best : NEG_;
    float4 bx; bx.x = xmin; bx.y = ymin; bx.z = xmax; bx.w = ymax;
    ((float4*)boxes)[g] = bx;
    meta[2 * g + 0] = (float)bi;
    meta[2 * g + 1] = best;
  }
}

// =====================================================================
// Kernel 2: sequential NMS, one 1024-thread block per batch image.
// Scores live in LDS (98 KB, OK on CDNA5's 320KB/WGP). 200 steps suffice:
// selected confs are non-increasing, so top_k(...,200) == first 200 steps.
// =====================================================================
__global__ __launch_bounds__(1024)
void nms_kernel(const float* __restrict__ scores,
                const float* __restrict__ boxes,
                const float* __restrict__ meta,
                float* __restrict__ out) {
  extern __shared__ unsigned char smem[];
  float* ssc   = (float*)smem;                 // N_ floats
  float* red_v = ssc + N_;                     // 32
  int*   red_i = (int*)(red_v + 32);           // 32
  float* sbox  = (float*)(red_i + 32);         // 4
  int*   sflag = (int*)(sbox + 4);             // [0]=valid, [1]=sel idx

  const int b   = blockIdx.x;
  const int tid = threadIdx.x;
  const int lane = tid & 31, wid = tid >> 5;

  const float*  srow = scores + (size_t)b * N_;
  const float4* brow = (const float4*)boxes + (size_t)b * N_;
  const float*  mrow = meta + (size_t)b * N_ * 2;
  float* outb = out + (size_t)b * (TOPK_ * 6);

  for (int i = tid; i < N_; i += 1024) ssc[i] = srow[i];
  __syncthreads();

  for (int k = 0; k < TOPK_; ++k) {
    // ---- block-wide first-max argmax over LDS scores ----
    float bv = -3.0e38f; int bi = 0x7FFFFFFF;
    for (int i = tid; i < N_; i += 1024) {
      float v = ssc[i];
      if (v > bv) { bv = v; bi = i; }   // indices increase -> first-max within thread
    }
#pragma unroll
    for (int m = 1; m < 32; m <<= 1) {  // wave32 xor-tree (ds permutes)
      float ov = __shfl_xor(bv, m, 32);
      int   oi = __shfl_xor(bi, m, 32);
      if (ov > bv || (ov == bv && oi < bi)) { bv = ov; bi = oi; }
    }
    if (lane == 0) { red_v[wid] = bv; red_i[wid] = bi; }
    __syncthreads();
    if (wid == 0) {
      bv = red_v[lane]; bi = red_i[lane];   // exactly 32 waves
#pragma unroll
      for (int m = 1; m < 32; m <<= 1) {
        float ov = __shfl_xor(bv, m, 32);
        int   oi = __shfl_xor(bi, m, 32);
        if (ov > bv || (ov == bv && oi < bi)) { bv = ov; bi = oi; }
      }
      if (lane == 0) {
        int valid = (bv > 0.5f * NEG_) ? 1 : 0;
        sflag[0] = valid; sflag[1] = bi;
        if (valid) {
          float4 sb = brow[bi];
          sbox[0] = sb.x; sbox[1] = sb.y; sbox[2] = sb.z; sbox[3] = sb.w;
          outb[k * 6 + 0] = mrow[2 * bi + 0];  // class_id
          outb[k * 6 + 1] = bv;                // conf
          outb[k * 6 + 2] = sb.x;
          outb[k * 6 + 3] = sb.y;
          outb[k * 6 + 4] = sb.z;
          outb[k * 6 + 5] = sb.w;
        }
      }
    }
    __syncthreads();

    if (!sflag[0]) {
      // no more valid selections: remaining rows are exactly zero
      for (int j = tid; j < (TOPK_ - k) * 6; j += 1024) outb[k * 6 + j] = 0.0f;
      return;
    }

    // ---- suppression: IoU of selected vs all (boxes served from L2) ----
    const int sel = sflag[1];
    const float sx1 = sbox[0], sy1 = sbox[1], sx2 = sbox[2], sy2 = sbox[3];
    const float sarea = (sx2 - sx1) * (sy2 - sy1);
    for (int i = tid; i < N_; i += 1024) {
      float4 bb = brow[i];
      float ix1 = fmaxf(sx1, bb.x), iy1 = fmaxf(sy1, bb.y);
      float ix2 = fminf(sx2, bb.z), iy2 = fminf(sy2, bb.w);
      float inter = fmaxf(ix2 - ix1, 0.0f) * fmaxf(iy2 - iy1, 0.0f);
      float area  = (bb.z - bb.x) * (bb.w - bb.y);
      float uni   = sarea + area - inter;
      float iou   = (uni > 0.0f) ? (inter / uni) : 0.0f;
      if (iou > IOU_T || i == sel) ssc[i] = NEG_;
    }
    __syncthreads();
  }
}

// =====================================================================
extern "C" void kernel_launch(void* const* d_in, const int* in_sizes, int n_in,
                              void* d_out, int out_size, void* d_ws, size_t ws_size,
                              hipStream_t stream) {
  (void)in_sizes; (void)n_in; (void)out_size; (void)ws_size;
  const float* y = (const float*)d_in[0];
  float* out = (float*)d_out;

  // workspace layout (floats): scores | boxes(float4) | meta(2f)  ~21 MB
  float* ws     = (float*)d_ws;
  float* scores = ws;                                  // B_*N_
  float* boxes  = scores + (size_t)B_ * N_;            // B_*N_*4 (16B-aligned offset)
  float* meta   = boxes + (size_t)B_ * N_ * 4;         // B_*N_*2

  const int tilesPerBatch = (N_ + 255) / 256;          // 96
  dim3 g1(B_ * tilesPerBatch), t1(256);
  size_t lds1 = (size_t)256 * F_ * sizeof(float);      // 95232 B
  decode_kernel<<<g1, t1, lds1, stream>>>(y, scores, boxes, meta);

  size_t lds2 = (size_t)N_ * sizeof(float) + 1024;     // ~99 KB
  nms_kernel<<<dim3(B_), dim3(1024), lds2, stream>>>(scores, boxes, meta, out);
}